// MultipCrossEntropySeesawLoss_52690658787514
// MI455X (gfx1250) — compile-verified
//
#include <hip/hip_runtime.h>
#include <math.h>
#include <stdint.h>

#define B_ROWS 8192
#define NCLS   2513
#define CTOT   2514          // NCLS + background column
#define P_EXP  0.8f
#define Q_EXP  2.0f
#define EPS_V  0.01f
#define NEG_BIG (-3.0e38f)   // avoid (-inf - -inf) = NaN in reductions

typedef float v2f __attribute__((ext_vector_type(2)));
typedef float v8f __attribute__((ext_vector_type(8)));

// ---------------- wave32 / block reductions (CDNA5 is wave32) ----------------
__device__ __forceinline__ float waveMax(float v) {
#pragma unroll
  for (int off = 16; off > 0; off >>= 1)
    v = fmaxf(v, __shfl_down(v, off, 32));
  return v;
}
__device__ __forceinline__ float waveSum(float v) {
#pragma unroll
  for (int off = 16; off > 0; off >>= 1)
    v += __shfl_down(v, off, 32);
  return v;
}
__device__ __forceinline__ float blockMax(float v, float* scr) {
  const int wid = threadIdx.x >> 5, lid = threadIdx.x & 31;
  v = waveMax(v);
  if (lid == 0) scr[wid] = v;
  __syncthreads();
  if (wid == 0) {
    float w = (lid < (int)(blockDim.x >> 5)) ? scr[lid] : NEG_BIG;
    w = waveMax(w);
    if (lid == 0) scr[0] = w;
  }
  __syncthreads();
  v = scr[0];
  __syncthreads();
  return v;
}
__device__ __forceinline__ float blockSum(float v, float* scr) {
  const int wid = threadIdx.x >> 5, lid = threadIdx.x & 31;
  v = waveSum(v);
  if (lid == 0) scr[wid] = v;
  __syncthreads();
  if (wid == 0) {
    float w = (lid < (int)(blockDim.x >> 5)) ? scr[lid] : 0.0f;
    w = waveSum(w);
    if (lid == 0) scr[0] = w;
  }
  __syncthreads();
  v = scr[0];
  __syncthreads();
  return v;
}

// ---------------- kernel 0: zero workspace accumulators ----------------
__global__ __launch_bounds__(256) void init_ws_kernel(float* __restrict__ counts,
                                                      int* __restrict__ labels,
                                                      int* __restrict__ validCnt) {
  const int i = blockIdx.x * blockDim.x + threadIdx.x;
  if (i < NCLS)   counts[i] = 0.0f;
  if (i < B_ROWS) labels[i] = 0;
  if (i == 0)     *validCnt = 0;
}

// ---------------- kernel 1: find one-hot label per row; class counts ----------------
// float2-vectorized streaming scan of target (rows are 8B-aligned: 2514*4 % 8 == 0).
// counts accumulate 1.0f values: integer-valued fp32 adds are exact -> order independent.
__global__ __launch_bounds__(256) void scan_target_kernel(const float2* __restrict__ target2,
                                                          int* __restrict__ labels,
                                                          float* __restrict__ counts,
                                                          int* __restrict__ validCnt) {
  const int cp  = blockIdx.x * blockDim.x + threadIdx.x;   // float2 index within row
  const int row = blockIdx.y;
  if (cp >= CTOT / 2) return;
  const float2 tv = target2[(size_t)row * (CTOT / 2) + cp];
  const int col0 = cp * 2;
#pragma unroll
  for (int k = 0; k < 2; ++k) {
    const float t  = (k == 0) ? tv.x : tv.y;
    const int col  = col0 + k;
    if (t > 0.5f && col > 0) {         // one-hot: at most one hit per row
      labels[row] = col;               // class index = col - 1
      atomicAdd(&counts[col - 1], 1.0f);
      atomicAdd(validCnt, 1);
    }
  }
}

// ---------------- kernel 2: lc[c] = log(max(cum_samples[c] + counts[c], 1)) ----------------
__global__ __launch_bounds__(256) void finalize_cum_kernel(const float* __restrict__ cum_samples,
                                                           const float* __restrict__ counts,
                                                           float* __restrict__ lc) {
  const int c = blockIdx.x * blockDim.x + threadIdx.x;
  if (c < NCLS) lc[c] = __logf(fmaxf(cum_samples[c] + counts[c], 1.0f));
}

// ---------------- kernel 3: per-row seesaw loss (one block per row) ----------------
__global__ __launch_bounds__(256) void seesaw_row_kernel(const float* __restrict__ input,
                                                         const int* __restrict__ labels,
                                                         const float* __restrict__ lc,
                                                         float* __restrict__ rowLoss) {
  __shared__ alignas(16) float srow[CTOT];   // full input row incl. bg col (10,056 B)
  __shared__ alignas(16) float slc[2516];    // staged log-cum table (10,056 B used)
  __shared__ float sred[8];
  const int row = blockIdx.x;
  const int lbl = labels[row];               // uniform across block
  if (lbl == 0) {                            // background / invalid row: contributes 0
    if (threadIdx.x == 0) rowLoss[row] = 0.0f;
    return;
  }

  // ---- async DMA global -> LDS (ASYNCcnt path, gfx1250): input row + lc table ----
  // input row byte offset = row*10056 (== 0 mod 8); lc is 256B-aligned in ws.
  {
    const unsigned long long grow =
        (unsigned long long)(uintptr_t)(const void*)(input + (size_t)row * CTOT);
    const unsigned long long glc =
        (unsigned long long)(uintptr_t)(const void*)lc;
    const unsigned int lrow = (unsigned int)(unsigned long long)(uintptr_t)(void*)&srow[0];
    const unsigned int llc  = (unsigned int)(unsigned long long)(uintptr_t)(void*)&slc[0];
    const int nch = (CTOT * 4) / 8;  // 1257 eight-byte chunks (covers 2514 floats exactly)
    for (int ch = threadIdx.x; ch < nch; ch += blockDim.x) {
      const unsigned int       off = (unsigned int)ch * 8u;
      const unsigned long long ga  = grow + off;
      const unsigned long long gb  = glc + off;   // reads into lc's padded region at tail (safe)
      const unsigned int       la  = lrow + off;
      const unsigned int       lb  = llc + off;
      asm volatile("global_load_async_to_lds_b64 %0, %1, off"
                   :: "v"(la), "v"(ga) : "memory");
      asm volatile("global_load_async_to_lds_b64 %0, %1, off"
                   :: "v"(lb), "v"(gb) : "memory");
    }
    asm volatile("s_wait_asynccnt 0" ::: "memory");  // this wave's async loads done
  }
  __syncthreads();                                    // all waves' LDS writes visible

  const float* x = &srow[1];  // x[j] = input[row, 1+j], j in [0, NCLS)

  // ---- pass A: rowmax and sumexp over x -> L1 = logsumexp(x) ----
  float lm = NEG_BIG;
  for (int j = threadIdx.x; j < NCLS; j += blockDim.x) lm = fmaxf(lm, x[j]);
  const float rowmax = blockMax(lm, sred);

  float ls = 0.0f;
  for (int j = threadIdx.x; j < NCLS; j += blockDim.x) ls += __expf(x[j] - rowmax);
  const float sumexp = blockSum(ls, sred);

  const float L1     = rowmax + __logf(sumexp);
  const float xstar  = srow[lbl];                     // x[j*], j* = lbl-1
  const float selfs  = __expf(xstar - L1);            // softmax score at target class
  const float ld     = __logf(fmaxf(selfs, EPS_V));   // log denom
  const float lcStar = slc[lbl - 1];
  __syncthreads();  // everyone done reading srow before in-place overwrite

  // ---- pass B: x2[j] = x[j] + log(mit) + log(comp) (0 at j*), stored in place ----
  float lm2 = NEG_BIG;
  for (int j = threadIdx.x; j < NCLS; j += blockDim.x) {
    const float xj = x[j];
    float x2;
    if (j == (lbl - 1)) {
      x2 = xj;                                        // (1 - t) kills the adjustment
    } else {
      const float lmit  = fminf(0.0f, P_EXP * (slc[j] - lcStar)); // log min(1,(cum_j/cum_*)^P)
      const float lcomp = Q_EXP * fmaxf(0.0f, (xj - L1) - ld);    // log max(1,ratio)^Q
      x2 = xj + lmit + lcomp;
    }
    srow[1 + j] = x2;   // same thread owns read & write slot
    lm2 = fmaxf(lm2, x2);
  }
  const float m2 = blockMax(lm2, sred);  // internal barriers publish srow writes

  float ls2 = 0.0f;
  for (int j = threadIdx.x; j < NCLS; j += blockDim.x) ls2 += __expf(x[j] - m2);
  const float s2 = blockSum(ls2, sred);

  if (threadIdx.x == 0)
    rowLoss[row] = (m2 + __logf(s2)) - xstar;         // -log_softmax(x2)[j*]
}

// ---------------- kernel 4: WMMA reduction of rowLoss (sum-with-ones GEMM) ----------------
// One wave32 (EXEC all-ones, as WMMA requires). D = A(16x4) * B_ones(4x16) + C accumulates
// 64 fp32 values per v_wmma_f32_16x16x4_f32; hardware accumulation order is fixed ->
// bitwise deterministic across replays. 8192 / 64 = 128 iterations.
__global__ __launch_bounds__(32) void finalize_loss_kernel(const float* __restrict__ rowLoss,
                                                           const int* __restrict__ validCnt,
                                                           float* __restrict__ out) {
  const int lane = threadIdx.x;   // 0..31, wave32
  v2f ones; ones[0] = 1.0f; ones[1] = 1.0f;
  v8f acc = {};
  const float* p = rowLoss + lane * 2;
  for (int it = 0; it < B_ROWS / 64; ++it) {
    v2f a;
    a[0] = p[it * 64 + 0];
    a[1] = p[it * 64 + 1];
    // 8 args: (neg_a, A, neg_b, B, c_mod, C, reuse_a, reuse_b)
    acc = __builtin_amdgcn_wmma_f32_16x16x4_f32(false, a, false, ones,
                                                (short)0, acc, false, false);
  }
  // acc[v] = D[M=v (+8 for lanes 16..31)][N=lane%16]; every column holds the row sums.
  float partial = acc[0] + acc[1] + acc[2] + acc[3] + acc[4] + acc[5] + acc[6] + acc[7];
  partial += __shfl_xor(partial, 16, 32);   // rows 0..7 + rows 8..15 -> full sum
  if (lane == 0) {
    const int cnt = *validCnt;
    out[0] = (cnt > 0) ? (partial / (float)cnt) : 0.0f;
  }
}

// ---------------- host-side launcher ----------------
extern "C" void kernel_launch(void* const* d_in, const int* in_sizes, int n_in,
                              void* d_out, int out_size, void* d_ws, size_t ws_size,
                              hipStream_t stream) {
  (void)in_sizes; (void)n_in; (void)out_size; (void)ws_size;
  const float* input       = (const float*)d_in[0];  // (B, CTOT) f32
  const float* target      = (const float*)d_in[1];  // (B, CTOT) f32 one-hot
  const float* cum_samples = (const float*)d_in[2];  // (NCLS,)   f32
  float* out = (float*)d_out;                        // scalar f32

  // workspace carve-up with 8B/256B-aligned, padded regions (~86 KB total)
  float* wsf     = (float*)d_ws;
  float* counts  = wsf + 0;        // 2513 used, padded to 2560
  float* lc      = wsf + 2560;     // 2513 used, padded to 2560 (256B-aligned for b64 DMA)
  float* rowLoss = wsf + 5120;     // 8192
  int*   labels   = (int*)(wsf + 13312); // 8192
  int*   validCnt = (int*)(wsf + 21504); // 1

  init_ws_kernel<<<(B_ROWS + 255) / 256, 256, 0, stream>>>(counts, labels, validCnt);

  dim3 g1((CTOT / 2 + 255) / 256, B_ROWS);
  scan_target_kernel<<<g1, 256, 0, stream>>>((const float2*)target, labels, counts, validCnt);

  finalize_cum_kernel<<<(NCLS + 255) / 256, 256, 0, stream>>>(cum_samples, counts, lc);

  seesaw_row_kernel<<<B_ROWS, 256, 0, stream>>>(input, labels, lc, rowLoss);

  finalize_loss_kernel<<<1, 32, 0, stream>>>(rowLoss, validCnt, out);
}